// GPTSanJapaneseTop1Router_55327768708082
// MI455X (gfx1250) — compile-verified
//
#include <hip/hip_runtime.h>

typedef __attribute__((ext_vector_type(2))) float v2f;
typedef __attribute__((ext_vector_type(8))) float v8f;

#define NUM_EXPERTS     64
#define EXPERT_CAPACITY 160
#define HIDDEN          2048
#define GROUPS          4
#define TOKENS          8192
#define TOTAL_TOKENS    (GROUPS * TOKENS)

// ---------------------------------------------------------------------------
// Kernel 1: router GEMM (f32 WMMA 16x16x4) + bias + softmax-max/argmax/sumexp
//   block = 128 threads (4 waves). Each wave: 32 tokens (2 M-tiles) x 64
//   experts (4 N-tiles). Block covers 128 tokens.
// ---------------------------------------------------------------------------
__global__ __launch_bounds__(128)
void router_gemm_kernel(const float* __restrict__ hidden,   // [G*T, H]
                        const float* __restrict__ W,        // [E, H]
                        const float* __restrict__ bias,     // [E]
                        float* __restrict__ out,            // full d_out
                        int* __restrict__ choice)           // [G*T] workspace
{
    __shared__ float lds_logits[128 * NUM_EXPERTS];         // 32 KB

    const int lane = threadIdx.x & 31;
    const int wave = threadIdx.x >> 5;
    const int m    = lane & 15;           // A-row within tile / B-col within tile
    const int khalf = lane >> 4;          // 0: K=0,1   1: K=2,3
    const int koff  = khalf << 1;

    const int tok_base = blockIdx.x * 128 + wave * 32;

    // A row pointers (2 M-tiles), each lane reads a contiguous K-pair
    const float* arow0 = hidden + (size_t)(tok_base +      m) * HIDDEN + koff;
    const float* arow1 = hidden + (size_t)(tok_base + 16 + m) * HIDDEN + koff;
    // B "column" pointers: B[k][n] = W[n][k]; lane's col n = ntile*16 + m
    const float* brow0 = W + (size_t)( 0 + m) * HIDDEN + koff;
    const float* brow1 = W + (size_t)(16 + m) * HIDDEN + koff;
    const float* brow2 = W + (size_t)(32 + m) * HIDDEN + koff;
    const float* brow3 = W + (size_t)(48 + m) * HIDDEN + koff;

    v8f acc[2][4];
#pragma unroll
    for (int mt = 0; mt < 2; ++mt)
#pragma unroll
        for (int n = 0; n < 4; ++n)
            acc[mt][n] = (v8f){0.f, 0.f, 0.f, 0.f, 0.f, 0.f, 0.f, 0.f};

#pragma unroll 2
    for (int kk = 0; kk < HIDDEN; kk += 4) {
        v2f a0 = *(const v2f*)(arow0 + kk);
        v2f a1 = *(const v2f*)(arow1 + kk);
        v2f b0 = *(const v2f*)(brow0 + kk);
        v2f b1 = *(const v2f*)(brow1 + kk);
        v2f b2 = *(const v2f*)(brow2 + kk);
        v2f b3 = *(const v2f*)(brow3 + kk);

        acc[0][0] = __builtin_amdgcn_wmma_f32_16x16x4_f32(false, a0, false, b0, (short)0, acc[0][0], false, false);
        acc[0][1] = __builtin_amdgcn_wmma_f32_16x16x4_f32(false, a0, false, b1, (short)0, acc[0][1], false, false);
        acc[0][2] = __builtin_amdgcn_wmma_f32_16x16x4_f32(false, a0, false, b2, (short)0, acc[0][2], false, false);
        acc[0][3] = __builtin_amdgcn_wmma_f32_16x16x4_f32(false, a0, false, b3, (short)0, acc[0][3], false, false);
        acc[1][0] = __builtin_amdgcn_wmma_f32_16x16x4_f32(false, a1, false, b0, (short)0, acc[1][0], false, false);
        acc[1][1] = __builtin_amdgcn_wmma_f32_16x16x4_f32(false, a1, false, b1, (short)0, acc[1][1], false, false);
        acc[1][2] = __builtin_amdgcn_wmma_f32_16x16x4_f32(false, a1, false, b2, (short)0, acc[1][2], false, false);
        acc[1][3] = __builtin_amdgcn_wmma_f32_16x16x4_f32(false, a1, false, b3, (short)0, acc[1][3], false, false);
    }

    // D layout (16x16 f32): VGPR v -> row v (lanes 0-15) / row v+8 (lanes 16-31),
    // col = lane & 15. Stage into LDS with bias added.
#pragma unroll
    for (int mt = 0; mt < 2; ++mt) {
#pragma unroll
        for (int n = 0; n < 4; ++n) {
            const int col = n * 16 + m;
            const float be = bias[col];
#pragma unroll
            for (int v = 0; v < 8; ++v) {
                const int row = v + (khalf << 3);
                lds_logits[(wave * 32 + mt * 16 + row) * NUM_EXPERTS + col] =
                    acc[mt][n][v] + be;
            }
        }
    }
    __syncthreads();

    // One thread per token: max / argmax / sumexp; emit logits, probs_max, choice
    {
        const int tf = blockIdx.x * 128 + threadIdx.x;      // flat token index
        const float* lg = &lds_logits[threadIdx.x * NUM_EXPERTS];

        float mx = lg[0];
        int   arg = 0;
#pragma unroll 8
        for (int e = 1; e < NUM_EXPERTS; ++e) {
            float v = lg[e];
            if (v > mx) { mx = v; arg = e; }
        }
        float s = 0.f;
#pragma unroll 8
        for (int e = 0; e < NUM_EXPERTS; ++e)
            s += __expf(lg[e] - mx);

        float* pmax_out   = out + (size_t)TOTAL_TOKENS * NUM_EXPERTS;
        float* logits_out = pmax_out + TOTAL_TOKENS;

        float* lrow = logits_out + (size_t)tf * NUM_EXPERTS;
#pragma unroll 4
        for (int e = 0; e < NUM_EXPERTS; e += 4) {
            float4 v4 = make_float4(lg[e], lg[e + 1], lg[e + 2], lg[e + 3]);
            *(float4*)(lrow + e) = v4;
        }
        pmax_out[tf] = 1.0f / s;     // max softmax prob = exp(0)/sumexp
        choice[tf]   = arg;
    }
}

// ---------------------------------------------------------------------------
// Kernel 2: per-(group, expert) inclusive cumsum over tokens + capacity mask.
//   One block per group, one thread per expert; choices staged in LDS.
// ---------------------------------------------------------------------------
__global__ __launch_bounds__(64)
void capacity_kernel(const int* __restrict__ choice,        // [G*T]
                     float* __restrict__ expert_index)      // [G*T, E] floats
{
    __shared__ int lds_choice[TOKENS];                      // 32 KB
    const int g = blockIdx.x;
    const int e = threadIdx.x;

    for (int i = e; i < TOKENS; i += 64)
        lds_choice[i] = choice[g * TOKENS + i];
    __syncthreads();

    int count = 0;
    size_t base = (size_t)g * TOKENS * NUM_EXPERTS + e;
    for (int t = 0; t < TOKENS; ++t) {
        const int one = (lds_choice[t] == e) ? 1 : 0;
        count += one;
        expert_index[base + (size_t)t * NUM_EXPERTS] =
            (one && (count <= EXPERT_CAPACITY)) ? 1.0f : 0.0f;
    }
}

// ---------------------------------------------------------------------------
extern "C" void kernel_launch(void* const* d_in, const int* in_sizes, int n_in,
                              void* d_out, int out_size, void* d_ws, size_t ws_size,
                              hipStream_t stream) {
    const float* hidden = (const float*)d_in[0];   // [4, 8192, 2048] f32
    const float* W      = (const float*)d_in[1];   // [64, 2048] f32
    const float* b      = (const float*)d_in[2];   // [64] f32
    float* out = (float*)d_out;
    int*   choice = (int*)d_ws;                    // 32768 ints = 128 KB scratch

    router_gemm_kernel<<<TOTAL_TOKENS / 128, 128, 0, stream>>>(hidden, W, b, out, choice);
    capacity_kernel<<<GROUPS, 64, 0, stream>>>(choice, out);
}